// Model_83141976916166
// MI455X (gfx1250) — compile-verified
//
#include <hip/hip_runtime.h>

typedef __attribute__((ext_vector_type(2))) float v2f;
typedef __attribute__((ext_vector_type(8))) float v8f;

#define BDIM    8
#define SEQ     4096
#define HN      16
#define PDIM    64
#define NDIM    64
#define SEG_LEN 512
#define NSEG    (SEQ / SEG_LEN)
#define TBLK    32
#define LDX     72                 // padded LDS row stride (floats); 288B rows keep 16B alignment
#define TILE_F  (TBLK * LDX)       // floats per LDS tile buffer

__global__ void __launch_bounds__(256) zero_out_kernel(float* __restrict__ p, int n) {
    int i = blockIdx.x * blockDim.x + threadIdx.x;
    if (i < n) p[i] = 0.0f;
}

__global__ void __launch_bounds__(256) mamba_states_kernel(
    const float* __restrict__ X,   // (b, s, h, p)
    const float* __restrict__ A,   // (b, s, h)
    const float* __restrict__ Bm,  // (b, s, h, n)
    float* __restrict__ out)       // (b, h, p, n), pre-zeroed; accumulated via atomics
{
    __shared__ float sX[2 * TILE_F];     // double-buffered weighted-X tiles
    __shared__ float sB[2 * TILE_F];     // double-buffered B tiles
    __shared__ float sAval[SEG_LEN];
    __shared__ float sW[SEG_LEN];
    __shared__ float sScan[256];

    const int tid = threadIdx.x;
    const int blk = blockIdx.x;
    const int bh  = blk % (BDIM * HN);
    const int seg = blk / (BDIM * HN);
    const int bb  = bh / HN;
    const int hh  = bh % HN;
    const int t0  = seg * SEG_LEN;
    const int t1  = t0 + SEG_LEN;

    // Base pointers for this (b, h): element (t, j) lives at base + t*1024 + j
    const float* Xp = X  + ((size_t)bb * SEQ * HN + hh) * (size_t)PDIM;
    const float* Bp = Bm + ((size_t)bb * SEQ * HN + hh) * (size_t)NDIM;
    const float* Ap = A  +  (size_t)bb * SEQ * HN + hh;          // stride HN per t

    // LDS byte offsets (low 32 bits of a shared-aperture address are the LDS offset) and SGPR bases
    const unsigned ldsX = (unsigned)(size_t)(void*)&sX[0];
    const unsigned ldsB = (unsigned)(size_t)(void*)&sB[0];
    const unsigned long long gX = (unsigned long long)(size_t)Xp;
    const unsigned long long gB = (unsigned long long)(size_t)Bp;

    // Async DMA of one TBLK x 64 tile of X and B straight into LDS (no VGPR staging).
    auto issue_tile = [&](int kb, int buf) {
#pragma unroll
        for (int it = 0; it < 2; ++it) {
            int lin = it * 256 + tid;                       // 0..511 float4 chunks
            int tt  = lin >> 4;                             // 0..31 (timestep in tile)
            int p4  = (lin & 15) << 2;                      // 0,4,...,60
            unsigned goff = (((unsigned)(t0 + kb + tt)) * 1024u + (unsigned)p4) * 4u;
            unsigned loff = (unsigned)((buf * TILE_F + tt * LDX + p4) * 4);
            unsigned dx = ldsX + loff;
            unsigned db = ldsB + loff;
            asm volatile("global_load_async_to_lds_b128 %0, %1, %2"
                         :: "v"(dx), "v"(goff), "s"(gX) : "memory");
            asm volatile("global_load_async_to_lds_b128 %0, %1, %2"
                         :: "v"(db), "v"(goff), "s"(gB) : "memory");
        }
    };

    // Kick off tile 0 DMA immediately; its latency hides behind the A-scan below.
    issue_tile(0, 0);

    // ---- 1) tail = sum_{t in [t1, SEQ)} A[b,t,h]  (cross-segment decay, log space)
    float part = 0.0f;
    for (int t = t1 + tid; t < SEQ; t += 256) part += Ap[(size_t)t * HN];
    sScan[tid] = part;
    __syncthreads();
    for (int off = 128; off > 0; off >>= 1) {
        float v = 0.0f;
        if (tid < off) v = sScan[tid] + sScan[tid + off];
        __syncthreads();
        if (tid < off) sScan[tid] = v;
        __syncthreads();
    }
    const float tail = sScan[0];
    __syncthreads();

    // ---- 2) load this segment's A values
    sAval[tid]       = Ap[(size_t)(t0 + tid) * HN];
    sAval[tid + 256] = Ap[(size_t)(t0 + tid + 256) * HN];
    __syncthreads();

    // ---- 3) parallel suffix scan -> per-timestep weights w[t] = exp(sum_{u>t} A[u])
    const float a0 = sAval[2 * tid];
    const float a1 = sAval[2 * tid + 1];
    sScan[tid] = a0 + a1;
    __syncthreads();
    for (int off = 1; off < 256; off <<= 1) {
        float v = sScan[tid];
        float w = (tid + off < 256) ? sScan[tid + off] : 0.0f;
        __syncthreads();
        sScan[tid] = v + w;
        __syncthreads();
    }
    const float sufNext = (tid < 255) ? sScan[tid + 1] : 0.0f;
    sW[2 * tid + 1] = expf(tail + sufNext);
    sW[2 * tid]     = expf(tail + sufNext + a1);

    // Tile 0 must be resident before the first compute pass.
    asm volatile("s_wait_asynccnt 0x0" ::: "memory");
    __syncthreads();

    // ---- 4) main GEMM: C[p,n] += sum_t (w[t]*X[t,p]) * B[t,n] via v_wmma_f32_16x16x4_f32
    const int wave = tid >> 5;
    const int lane = tid & 31;
    const int half = lane >> 4;          // 0: lanes 0-15, 1: lanes 16-31
    const int l16  = lane & 15;
    const int mi   = wave >> 1;          // row tile 0..3
    const int ni0  = (wave & 1) * 2;     // first of two col tiles
    const int pA   = mi * 16 + l16;      // A-matrix row (= p)
    const int nB0  = ni0 * 16 + l16;     // B-matrix col (= n)
    const int koff = half * 2;           // K offset per half-wave (ISA 16x4 layout)

    v8f acc0 = {};
    v8f acc1 = {};

    int cur = 0;
    for (int kb = 0; kb < SEG_LEN; kb += TBLK) {
        // Prefetch next tile into the other buffer while computing on this one.
        if (kb + TBLK < SEG_LEN) issue_tile(kb + TBLK, cur ^ 1);

        const float* bx = sX + cur * TILE_F;
        const float* bn = sB + cur * TILE_F;
        const float* wp = sW + kb;

#pragma unroll
        for (int kk = 0; kk < TBLK; kk += 4) {
            const int r0 = kk + koff;               // even -> sW pair is 8B aligned
            const float w0 = wp[r0];
            const float w1 = wp[r0 + 1];
            v2f af, bf0, bf1;
            af[0]  = bx[r0 * LDX + pA] * w0;        // fold decay into A rows
            af[1]  = bx[(r0 + 1) * LDX + pA] * w1;
            bf0[0] = bn[r0 * LDX + nB0];
            bf0[1] = bn[(r0 + 1) * LDX + nB0];
            bf1[0] = bn[r0 * LDX + nB0 + 16];
            bf1[1] = bn[(r0 + 1) * LDX + nB0 + 16];
            acc0 = __builtin_amdgcn_wmma_f32_16x16x4_f32(
                false, af, false, bf0, (short)0, acc0, false, false);
            acc1 = __builtin_amdgcn_wmma_f32_16x16x4_f32(
                false, af, false, bf1, (short)0, acc1, false, false);
        }

        // All of this wave's async writes to buf^1 done, then block-wide handoff.
        asm volatile("s_wait_asynccnt 0x0" ::: "memory");
        __syncthreads();
        cur ^= 1;
    }

    // ---- 5) accumulate into out (b,h,p,n); C/D layout: VGPR r -> row r (+8 upper half-wave)
    float* outp = out + (size_t)(bb * HN + hh) * (PDIM * NDIM);
#pragma unroll
    for (int r = 0; r < 8; ++r) {
        int row = mi * 16 + r + half * 8;
        atomicAdd(outp + row * NDIM + nB0,      acc0[r]);
        atomicAdd(outp + row * NDIM + nB0 + 16, acc1[r]);
    }
}

extern "C" void kernel_launch(void* const* d_in, const int* in_sizes, int n_in,
                              void* d_out, int out_size, void* d_ws, size_t ws_size,
                              hipStream_t stream) {
    const float* X = (const float*)d_in[0];   // (8, 4096, 16, 64)
    const float* A = (const float*)d_in[1];   // (8, 4096, 16)
    const float* B = (const float*)d_in[2];   // (8, 4096, 16, 64)
    float* out = (float*)d_out;               // (8, 16, 64, 64)

    zero_out_kernel<<<(out_size + 255) / 256, 256, 0, stream>>>(out, out_size);
    mamba_states_kernel<<<BDIM * HN * NSEG, 256, 0, stream>>>(X, A, B, out);
}